// SimpleRecGNN_87247965651115
// MI455X (gfx1250) — compile-verified
//
#include <hip/hip_runtime.h>

#define NUM_USERS 100000
#define NUM_BOOKS 100000

typedef __attribute__((ext_vector_type(2))) float v2f;
typedef __attribute__((ext_vector_type(8))) float v8f;

// ---------------- degree / normalization ----------------

__global__ void k_init_deg(float* __restrict__ deg, int n) {
    int i = blockIdx.x * blockDim.x + threadIdx.x;
    if (i < n) deg[i] = 1.0f;              // self-loop weight
}

__global__ void k_accum_deg(const int* __restrict__ col, const float* __restrict__ w,
                            float* __restrict__ deg, int e) {
    int i = blockIdx.x * blockDim.x + threadIdx.x;
    if (i < e) atomicAdd(&deg[col[i]], w[i]);
}

__global__ void k_dinv(float* __restrict__ deg, int n) {
    int i = blockIdx.x * blockDim.x + threadIdx.x;
    if (i < n) {
        float d = deg[i];
        deg[i] = (d > 0.0f) ? rsqrtf(fmaxf(d, 1e-12f)) : 0.0f;   // in-place deg -> dinv
    }
}

// ---------------- initial embedding gather ----------------

__global__ void k_gather(const int* __restrict__ x, const float* __restrict__ ue,
                         const float* __restrict__ be, float* __restrict__ h0, int n) {
    int t = blockIdx.x * blockDim.x + threadIdx.x;     // one thread per (node, 4-float chunk)
    if (t >= n * 4) return;
    int node = t >> 2, c = t & 3;
    int idx = x[node * 2];
    int typ = x[node * 2 + 1];
    const float4* src;
    if (typ == 0) {
        int ui = min(max(idx, 0), NUM_USERS - 1);
        src = (const float4*)(ue + (size_t)ui * 16);
    } else {
        int bi = min(max(idx - NUM_USERS, 0), NUM_BOOKS - 1);
        src = (const float4*)(be + (size_t)bi * 16);
    }
    ((float4*)(h0 + (size_t)node * 16))[c] = src[c];
}

// ---------------- dense transform: one wave -> 16x16 output tile via WMMA ----------------
// Compile-time shape + ReLU flag: fully unrolled K-loop, no per-element selects.

template <int DIN, int DOUT, bool RELU>
__global__ void k_gemm_wmma(const float* __restrict__ hin, const float* __restrict__ W,
                            float* __restrict__ hout, int n) {
    const int lane = threadIdx.x;          // 0..31, wave32
    const int m    = lane & 15;
    const int hi   = lane >> 4;            // 0: K0/K1 half, 1: K2/K3 half
    const int row0 = blockIdx.x * 16;
    const int ncol = blockIdx.y * 16 + m;  // B-matrix column for this lane
    const int row  = min(row0 + m, n - 1); // clamped A-row (store side is guarded)

    v8f acc = {};
#pragma unroll
    for (int k0 = 0; k0 < DIN; k0 += 4) {
        const int ka = k0 + hi * 2;
        float a0 = hin[(size_t)row * DIN + ka];
        float a1 = hin[(size_t)row * DIN + ka + 1];
        if (RELU) { a0 = fmaxf(a0, 0.0f); a1 = fmaxf(a1, 0.0f); }  // fused ReLU of prev layer
        v2f a; a.x = a0; a.y = a1;
        v2f b; b.x = W[ka * DOUT + ncol];
               b.y = W[(ka + 1) * DOUT + ncol];
        // D = A(16x4) * B(4x16) + C   -> v_wmma_f32_16x16x4_f32
        acc = __builtin_amdgcn_wmma_f32_16x16x4_f32(
            /*neg_a=*/false, a, /*neg_b=*/false, b,
            /*c_mod=*/(short)0, acc, /*reuse_a=*/false, /*reuse_b=*/false);
    }

    // D layout: VGPR v, lanes 0-15 -> (M=v, N=lane); lanes 16-31 -> (M=v+8, N=lane-16).
    // Rows within the 8-store column differ by 1 -> byte stride DOUT*4: one base pointer,
    // immediate-offset stores (wave-uniform fast path; N%16==0 here so it always runs).
    if (row0 + 16 <= n) {
        float* p = hout + (size_t)(row0 + hi * 8) * DOUT + blockIdx.y * 16 + m;
#pragma unroll
        for (int v = 0; v < 8; ++v) p[(size_t)v * DOUT] = acc[v];
    } else {
#pragma unroll
        for (int v = 0; v < 8; ++v) {
            int r = row0 + hi * 8 + v;
            if (r < n) hout[(size_t)r * DOUT + blockIdx.y * 16 + m] = acc[v];
        }
    }
}

// ---------------- self-loop + bias init of accumulator ----------------

template <int DOUT>
__global__ void k_init_agg(const float* __restrict__ hw, const float* __restrict__ dinv,
                           const float* __restrict__ bias, float* __restrict__ agg,
                           int total) {
    int t = blockIdx.x * blockDim.x + threadIdx.x;
    if (t >= total) return;
    int node = t / DOUT;                   // constant power-of-two -> shift
    int c    = t % DOUT;
    float di = dinv[node];
    agg[t] = hw[t] * di * di + bias[c];    // self-loop norm = dinv^2, w=1
}

// ---------------- edge scatter: lane per (edge, feature), L2-resident f32 atomics -----------

template <int DOUT>
__global__ void k_scatter(const float* __restrict__ hw, const float* __restrict__ dinv,
                          const int* __restrict__ row, const int* __restrict__ col,
                          const float* __restrict__ w, float* __restrict__ agg,
                          int total) {
    int t = blockIdx.x * blockDim.x + threadIdx.x;
    if (t >= total) return;
    int ed = t / DOUT;                     // DOUT=32 -> whole wave shares one edge (uniform loads)
    int c  = t % DOUT;
    int r  = row[ed];
    int cl = col[ed];
    float nrm = dinv[r] * w[ed] * dinv[cl];
    atomicAdd(&agg[(size_t)cl * DOUT + c], hw[(size_t)r * DOUT + c] * nrm);
}

// ---------------- per-layer launch helper ----------------

template <int DIN, int DOUT, bool RELU>
static void run_layer(const float* hin, const float* W, const float* b,
                      const float* dinv, float* hw, float* agg,
                      const int* rowv, const int* colv, const float* ew,
                      int n, int e, hipStream_t stream) {
    dim3 ggrid((n + 15) / 16, DOUT / 16);
    k_gemm_wmma<DIN, DOUT, RELU><<<ggrid, 32, 0, stream>>>(hin, W, hw, n);

    const int tot1 = n * DOUT;
    k_init_agg<DOUT><<<(tot1 + 255) / 256, 256, 0, stream>>>(hw, dinv, b, agg, tot1);

    const long tot2 = (long)e * DOUT;      // <= 64M, fits int
    k_scatter<DOUT><<<(int)((tot2 + 255) / 256), 256, 0, stream>>>(hw, dinv, rowv, colv, ew,
                                                                   agg, (int)tot2);
}

// ---------------- driver ----------------

extern "C" void kernel_launch(void* const* d_in, const int* in_sizes, int n_in,
                              void* d_out, int out_size, void* d_ws, size_t ws_size,
                              hipStream_t stream) {
    const int*   x  = (const int*)d_in[0];
    const int*   ei = (const int*)d_in[1];
    const float* ew = (const float*)d_in[2];
    const float* ue = (const float*)d_in[3];
    const float* be = (const float*)d_in[4];
    const float* Ws[6]; const float* bs[6];
    for (int i = 0; i < 6; ++i) {
        Ws[i] = (const float*)d_in[5 + 2 * i];
        bs[i] = (const float*)d_in[6 + 2 * i];
    }
    const int n = in_sizes[0] / 2;         // 200000
    const int e = in_sizes[2];             // 2000000
    const int* rowv = ei;
    const int* colv = ei + e;

    // workspace: dinv[N] | hA[N*32] | hB[N*32] | hw[N*32]  (~78 MB)
    float* dinv = (float*)d_ws;
    float* hA   = dinv + n;
    float* hB   = hA + (size_t)n * 32;
    float* hw   = hB + (size_t)n * 32;

    k_init_deg <<<(n + 255) / 256, 256, 0, stream>>>(dinv, n);
    k_accum_deg<<<(e + 255) / 256, 256, 0, stream>>>(colv, ew, dinv, e);
    k_dinv     <<<(n + 255) / 256, 256, 0, stream>>>(dinv, n);
    k_gather   <<<(n * 4 + 255) / 256, 256, 0, stream>>>(x, ue, be, hA, n);

    float* out = (float*)d_out;

    // layer 0: 16 -> 32, no input ReLU (reads raw embeddings)
    run_layer<16, 32, false>(hA, Ws[0], bs[0], dinv, hw, hB, rowv, colv, ew, n, e, stream);
    // layers 1-4: 32 -> 32, ReLU fused into A-load
    run_layer<32, 32, true >(hB, Ws[1], bs[1], dinv, hw, hA, rowv, colv, ew, n, e, stream);
    run_layer<32, 32, true >(hA, Ws[2], bs[2], dinv, hw, hB, rowv, colv, ew, n, e, stream);
    run_layer<32, 32, true >(hB, Ws[3], bs[3], dinv, hw, hA, rowv, colv, ew, n, e, stream);
    run_layer<32, 32, true >(hA, Ws[4], bs[4], dinv, hw, hB, rowv, colv, ew, n, e, stream);
    // layer 5: 32 -> 16, ReLU on input, aggregate straight into d_out (no output ReLU)
    run_layer<32, 16, true >(hB, Ws[5], bs[5], dinv, hw, out, rowv, colv, ew, n, e, stream);
}